// GraphNN_61186104099485
// MI455X (gfx1250) — compile-verified
//
#include <hip/hip_runtime.h>

typedef __attribute__((ext_vector_type(2))) float v2f;
typedef __attribute__((ext_vector_type(8))) float v8f;

#define DFEAT 128

// ---------------------------------------------------------------------------
// Zero agg + deg scratch (they are laid out contiguously in d_ws).
// ---------------------------------------------------------------------------
__global__ __launch_bounds__(256) void zero_f32(float* __restrict__ p, int n) {
  int i = blockIdx.x * blockDim.x + threadIdx.x;
  if (i < n) p[i] = 0.0f;
}

// ---------------------------------------------------------------------------
// One wave (32 lanes) per edge: lane l moves features [4l, 4l+4) of x[src]
// into agg[dst] via global_atomic_add_f32 (no-return -> STOREcnt only).
// Everything lives in the 192MB L2, so this is L2-atomic-throughput bound.
// ---------------------------------------------------------------------------
__global__ __launch_bounds__(256) void scatter_mean_accum(
    const float* __restrict__ x, const int* __restrict__ src,
    const int* __restrict__ dst, float* __restrict__ agg,
    float* __restrict__ deg, int E) {
  int e    = (blockIdx.x * blockDim.x + threadIdx.x) >> 5;
  int lane = threadIdx.x & 31;
  if (e >= E) return;
  int s = src[e];                       // uniform per wave -> scalar load
  int d = dst[e];
  const float4 v = *reinterpret_cast<const float4*>(x + (size_t)s * DFEAT + lane * 4);
  float* a = agg + (size_t)d * DFEAT + lane * 4;
  atomicAdd(a + 0, v.x);
  atomicAdd(a + 1, v.y);
  atomicAdd(a + 2, v.z);
  atomicAdd(a + 3, v.w);
  if (lane == 0) atomicAdd(deg + d, 1.0f);
}

// ---------------------------------------------------------------------------
// out = X @ Wself^T + (AGG/deg) @ Wneigh^T + b  [+ ReLU]
// One wave per 16x16 tile of out; fp32 WMMA 16x16x4 (full precision).
// A layout (16x4 f32): lanes 0-15 -> K=k0,k0+1 ; lanes 16-31 -> K=k0+2,k0+3
// B layout (4x16 f32): VGPR0 holds K rows {k0, k0+2}, VGPR1 holds {k0+1, k0+3}
//   and B[k][n] = W[n][k], so both A and B are contiguous b64 loads.
// ---------------------------------------------------------------------------
template <bool RELU>
__global__ __launch_bounds__(256) void sage_transform(
    const float* __restrict__ X, const float* __restrict__ AGG,
    const float* __restrict__ DEG, const float* __restrict__ Wself,
    const float* __restrict__ Wneigh, const float* __restrict__ bias,
    float* __restrict__ OUT, int N) {
  const int lane    = threadIdx.x & 31;
  const int waveId  = threadIdx.x >> 5;      // 8 waves = 8 column tiles (8*16 == 128)
  const int rowBase = blockIdx.x * 16;
  int m = rowBase + (lane & 15);             // this lane's A row
  if (m >= N) m = N - 1;                     // clamp (N % 16 == 0 in practice)
  const int col   = waveId * 16 + (lane & 15);
  const int khalf = (lane >> 4) << 1;        // 0 or 2

  const float* __restrict__ arow = X + (size_t)m * DFEAT;
  const float* __restrict__ brow = Wself + (size_t)col * DFEAT;  // B[k][n] = W[n][k]
  v8f c = {};
#pragma unroll
  for (int k0 = 0; k0 < DFEAT; k0 += 4) {
    v2f a = *reinterpret_cast<const v2f*>(arow + k0 + khalf);
    v2f b = *reinterpret_cast<const v2f*>(brow + k0 + khalf);
    c = __builtin_amdgcn_wmma_f32_16x16x4_f32(false, a, false, b, (short)0, c,
                                              false, false);
  }

  // mean aggregation folded into the A-load of pass 2 (one rcp per lane-row)
  const float r = 1.0f / fmaxf(DEG[m], 1.0f);
  arow = AGG + (size_t)m * DFEAT;
  brow = Wneigh + (size_t)col * DFEAT;
#pragma unroll
  for (int k0 = 0; k0 < DFEAT; k0 += 4) {
    v2f a = *reinterpret_cast<const v2f*>(arow + k0 + khalf);
    a *= r;
    v2f b = *reinterpret_cast<const v2f*>(brow + k0 + khalf);
    c = __builtin_amdgcn_wmma_f32_16x16x4_f32(false, a, false, b, (short)0, c,
                                              false, false);
  }

  // epilogue: bias + optional ReLU; C/D layout: VGPR v -> row v + 8*(lane>=16)
  const float bb = bias[col];
#pragma unroll
  for (int v = 0; v < 8; ++v) {
    int row = rowBase + v + ((lane >> 4) << 3);
    if (row < N) {
      float val = c[v] + bb;
      if (RELU) val = fmaxf(val, 0.0f);
      OUT[(size_t)row * DFEAT + col] = val;
    }
  }
}

// ---------------------------------------------------------------------------
// inputs: 0 x [N,128] f32 | 1 edge_index [2,E] int | 2 W1_self | 3 W1_neigh
//         4 b1 | 5 W2_self | 6 W2_neigh | 7 b2        output: [N,128] f32
// ---------------------------------------------------------------------------
extern "C" void kernel_launch(void* const* d_in, const int* in_sizes, int n_in,
                              void* d_out, int out_size, void* d_ws, size_t ws_size,
                              hipStream_t stream) {
  const float* x   = (const float*)d_in[0];
  const int*   ei  = (const int*)d_in[1];
  const float* W1s = (const float*)d_in[2];
  const float* W1n = (const float*)d_in[3];
  const float* b1  = (const float*)d_in[4];
  const float* W2s = (const float*)d_in[5];
  const float* W2n = (const float*)d_in[6];
  const float* b2  = (const float*)d_in[7];
  float* out = (float*)d_out;

  const int N = in_sizes[0] / DFEAT;   // 10000
  const int E = in_sizes[1] / 2;       // 640000
  const int* src = ei;                 // edge_index[0]
  const int* dst = ei + E;             // edge_index[1]

  // workspace: agg [N*D] | deg [N] | h [N*D]   (~10.3 MB)
  float* agg = (float*)d_ws;
  float* deg = agg + (size_t)N * DFEAT;
  float* h   = deg + N;

  const int nZero         = N * DFEAT + N;          // agg+deg are contiguous
  const int zeroBlocks    = (nZero + 255) / 256;
  const int scatterBlocks = (E * 32 + 255) / 256;   // one wave per edge
  const int gemmBlocks    = (N + 15) / 16;          // 625 row tiles

  // ---- layer 1 ----
  zero_f32<<<zeroBlocks, 256, 0, stream>>>(agg, nZero);
  scatter_mean_accum<<<scatterBlocks, 256, 0, stream>>>(x, src, dst, agg, deg, E);
  sage_transform<true><<<gemmBlocks, 256, 0, stream>>>(x, agg, deg, W1s, W1n, b1, h, N);

  // ---- layer 2 ----
  zero_f32<<<zeroBlocks, 256, 0, stream>>>(agg, nZero);
  scatter_mean_accum<<<scatterBlocks, 256, 0, stream>>>(h, src, dst, agg, deg, E);
  sage_transform<false><<<gemmBlocks, 256, 0, stream>>>(h, agg, deg, W2s, W2n, b2, out, N);
}